// ManifoldSmoothingLayer_81003083202621
// MI455X (gfx1250) — compile-verified
//
#include <hip/hip_runtime.h>

typedef __attribute__((ext_vector_type(2))) float v2f;
typedef __attribute__((ext_vector_type(8))) float v8f;

#define NPTS 8192
#define DIM  512
#define KNN  10
#define ROWS 16
#define NWAVES 4
#define CTILES 4                                  // 16-col tiles per wave per sweep
#define COLS_PER_ITER (NWAVES * CTILES * 16)      // 256
#define NITER (NPTS / COLS_PER_ITER)              // 32

// ---------------- kernel 1: per-row squared norms --------------------------
__global__ __launch_bounds__(256) void sq_norm_kernel(const float* __restrict__ x,
                                                      float* __restrict__ sq) {
  const int wave = threadIdx.x >> 5;
  const int lane = threadIdx.x & 31;
  const int row  = blockIdx.x * 8 + wave;
  const float* xr = x + (size_t)row * DIM;
  float s = 0.f;
  #pragma unroll 4
  for (int t = lane; t < DIM; t += 32) {
    float v = xr[t];
    s = fmaf(v, v, s);
  }
  #pragma unroll
  for (int off = 16; off > 0; off >>= 1) s += __shfl_xor(s, off, 32);
  if (lane == 0) sq[row] = s;
}

// ------------- kernel 2: fused distance GEMM + top-10 + smoothing ----------
__global__ __launch_bounds__(128) void knn_smooth_kernel(const float* __restrict__ x,
                                                         const float* __restrict__ sq,
                                                         float* __restrict__ out) {
  __shared__ float Alds[DIM * ROWS];           // A panel, transposed [k][m]  (32 KB)
  __shared__ float tileS[NWAVES][ROWS * 17];   // per-wave score tile, padded
  __shared__ float cand_d[ROWS][NWAVES * 2 * KNN];
  __shared__ int   cand_i[ROWS][NWAVES * 2 * KNN];
  __shared__ int   selIdx[ROWS][KNN];

  const int tid  = threadIdx.x;
  const int wave = tid >> 5;
  const int lane = tid & 31;
  const int half = lane >> 4;   // 0: lanes 0-15, 1: lanes 16-31
  const int nn   = lane & 15;   // M for A-frag / N for B,C,D frags
  const int r0   = blockIdx.x * ROWS;

  // ---- stage the 16 A rows into LDS, transposed to k-major ----
  {
    const int m  = tid >> 3;    // 0..15 (row within block)
    const int t8 = tid & 7;     // 8 threads per row, 64 k's each
    const float* xr = x + (size_t)(r0 + m) * DIM + t8 * 64;
    #pragma unroll
    for (int kk = 0; kk < 64; kk += 4) {
      float4 v = *(const float4*)(xr + kk);
      const int kb = t8 * 64 + kk;
      Alds[(kb + 0) * ROWS + m] = v.x;
      Alds[(kb + 1) * ROWS + m] = v.y;
      Alds[(kb + 2) * ROWS + m] = v.z;
      Alds[(kb + 3) * ROWS + m] = v.w;
    }
  }
  __syncthreads();

  // private sorted top-10 (ascending; bd[9] is current cutoff)
  float bd[KNN];
  int   bi[KNN];
  #pragma unroll
  for (int s = 0; s < KNN; ++s) { bd[s] = 3.402823466e38f; bi[s] = 0; }

  // ---- sweep: each wave computes a 16x64 strip (4 accumulators) per iter ----
  for (int it = 0; it < NITER; ++it) {
    const int cg = (it * NWAVES + wave) * (CTILES * 16);

    // warm L0/L2 for next sweep's rows (global_prefetch_b8)
    if (it + 1 < NITER) {
      const int cgn = cg + COLS_PER_ITER;
      #pragma unroll
      for (int t = 0; t < CTILES; ++t)
        __builtin_prefetch(x + (size_t)(cgn + t * 16 + nn) * DIM, 0, 1);
    }

    const float* bp0 = x + (size_t)(cg +  0 + nn) * DIM;
    const float* bp1 = x + (size_t)(cg + 16 + nn) * DIM;
    const float* bp2 = x + (size_t)(cg + 32 + nn) * DIM;
    const float* bp3 = x + (size_t)(cg + 48 + nn) * DIM;

    v8f acc[CTILES];
    #pragma unroll
    for (int t = 0; t < CTILES; ++t)
      acc[t] = (v8f){0.f, 0.f, 0.f, 0.f, 0.f, 0.f, 0.f, 0.f};

    #pragma unroll 2
    for (int kk = 0; kk < DIM; kk += 4) {
      const int k0 = kk + 2 * half;          // ISA A/B frag: half selects K pair
      v2f a;
      a.x = Alds[(k0 + 0) * ROWS + nn];      // A[m=nn][k0]
      a.y = Alds[(k0 + 1) * ROWS + nn];      // A[m=nn][k0+1]
      v2f b0 = *(const v2f*)(bp0 + k0);
      v2f b1 = *(const v2f*)(bp1 + k0);
      v2f b2 = *(const v2f*)(bp2 + k0);
      v2f b3 = *(const v2f*)(bp3 + k0);
      acc[0] = __builtin_amdgcn_wmma_f32_16x16x4_f32(false, a, false, b0,
                                                     (short)0, acc[0], false, false);
      acc[1] = __builtin_amdgcn_wmma_f32_16x16x4_f32(false, a, false, b1,
                                                     (short)0, acc[1], false, false);
      acc[2] = __builtin_amdgcn_wmma_f32_16x16x4_f32(false, a, false, b2,
                                                     (short)0, acc[2], false, false);
      acc[3] = __builtin_amdgcn_wmma_f32_16x16x4_f32(false, a, false, b3,
                                                     (short)0, acc[3], false, false);
    }

    // ---- epilogue: 4 tiles, wave-private LDS bounce; same-wave DS ops are
    //      in-order (DScnt), so no block barrier is needed here. ----
    #pragma unroll
    for (int t = 0; t < CTILES; ++t) {
      const int cb0 = cg + t * 16;
      const float sqb = sq[cb0 + nn];
      // score[m][n] = sq[n] - 2*dot  (row term sq[m] is rank-invariant, dropped)
      #pragma unroll
      for (int i = 0; i < 8; ++i) {
        const int m = i + 8 * half;          // C/D layout: vgpr i -> M=i (+8 hi half)
        tileS[wave][m * 17 + nn] = fmaf(-2.f, acc[t][i], sqb);
      }
      // each lane scans 8 values of one row, maintains private top-10
      const int r  = nn;
      const int jb = half * 8;
      #pragma unroll
      for (int j = 0; j < 8; ++j) {
        const float v  = tileS[wave][r * 17 + jb + j];
        const int   ci = cb0 + jb + j;
        if (v < bd[KNN - 1]) {               // rare path after warmup
          bd[KNN - 1] = v;
          bi[KNN - 1] = ci;
          #pragma unroll
          for (int s = KNN - 1; s > 0; --s) {
            if (bd[s] < bd[s - 1]) {
              float td = bd[s]; bd[s] = bd[s - 1]; bd[s - 1] = td;
              int   ti = bi[s]; bi[s] = bi[s - 1]; bi[s - 1] = ti;
            }
          }
        }
      }
    }
  }

  // ---- publish the 8 partial lists per row ----
  {
    const int listIdx = wave * 2 + half;     // 0..7
    #pragma unroll
    for (int s = 0; s < KNN; ++s) {
      cand_d[nn][listIdx * KNN + s] = bd[s];
      cand_i[nn][listIdx * KNN + s] = bi[s];
    }
  }
  __syncthreads();

  // ---- one thread per row merges 80 candidates -> final 10 ----
  if (tid < ROWS) {
    float fd[KNN]; int fi[KNN];
    #pragma unroll
    for (int s = 0; s < KNN; ++s) { fd[s] = 3.402823466e38f; fi[s] = 0; }
    for (int c = 0; c < NWAVES * 2 * KNN; ++c) {
      const float v  = cand_d[tid][c];
      const int   ci = cand_i[tid][c];
      if (v < fd[KNN - 1]) {
        fd[KNN - 1] = v; fi[KNN - 1] = ci;
        #pragma unroll
        for (int s = KNN - 1; s > 0; --s) {
          if (fd[s] < fd[s - 1]) {
            float td = fd[s]; fd[s] = fd[s - 1]; fd[s - 1] = td;
            int   ti = fi[s]; fi[s] = fi[s - 1]; fi[s - 1] = ti;
          }
        }
      }
    }
    #pragma unroll
    for (int s = 0; s < KNN; ++s) selIdx[tid][s] = fi[s];
  }
  __syncthreads();

  // ---- smoothed output: 0.5*x + 0.05*sum(10 neighbors), float4 per thread ----
  const int cb = tid * 4;                    // 128 threads * 4 cols = 512
  for (int r = 0; r < ROWS; ++r) {
    float4 self = *(const float4*)(x + (size_t)(r0 + r) * DIM + cb);
    float sx = 0.f, sy = 0.f, sz = 0.f, sw = 0.f;
    #pragma unroll
    for (int t = 0; t < KNN; ++t) {
      const float* nr = x + (size_t)selIdx[r][t] * DIM + cb;
      float4 nv = *(const float4*)nr;
      sx += nv.x; sy += nv.y; sz += nv.z; sw += nv.w;
    }
    float4 o;
    o.x = 0.5f * self.x + 0.05f * sx;
    o.y = 0.5f * self.y + 0.05f * sy;
    o.z = 0.5f * self.z + 0.05f * sz;
    o.w = 0.5f * self.w + 0.05f * sw;
    *(float4*)(out + (size_t)(r0 + r) * DIM + cb) = o;
  }
}

extern "C" void kernel_launch(void* const* d_in, const int* in_sizes, int n_in,
                              void* d_out, int out_size, void* d_ws, size_t ws_size,
                              hipStream_t stream) {
  (void)in_sizes; (void)n_in; (void)out_size; (void)ws_size;
  const float* x  = (const float*)d_in[0];
  float* out      = (float*)d_out;
  float* sq       = (float*)d_ws;            // 8192 floats of scratch

  sq_norm_kernel<<<NPTS / 8, 256, 0, stream>>>(x, sq);
  knn_smooth_kernel<<<NPTS / ROWS, 128, 0, stream>>>(x, sq, out);
}